// I_Softmax_61289183314005
// MI455X (gfx1250) — compile-verified
//
#include <hip/hip_runtime.h>
#include <hip/hip_bf16.h>
#include <math.h>

#define B_ROWS   16384
#define C_COLS   8192
#define CHUNKS   64        /* B_ROWS / 256 */
#define THREADS  256

#define MARGIN_M 2.00001f  /* M + EPS */
#define MARGIN_N 0.2f

/* ---- CDNA5 async global->LDS path (guarded; fallback = plain b128 loads) ---- */
#if defined(__has_builtin)
#if __has_builtin(__builtin_amdgcn_global_load_async_to_lds_b128)
#define HAVE_ASYNC_B128 1
#endif
#if __has_builtin(__builtin_amdgcn_s_wait_asynccnt)
#define HAVE_WAIT_ASYNC 1
#endif
#endif

typedef int   int4v  __attribute__((ext_vector_type(4)));
typedef float fvec4  __attribute__((ext_vector_type(4)));
/* builtin signature (from clang diagnostic): param0 = int4 in AS(1) ("__device__"),
   param1 = int4 in AS(3), then imm offset, imm cpol */
#define AS1V(p) ((__attribute__((address_space(1))) int4v*)(p))
#define AS3V(p) ((__attribute__((address_space(3))) int4v*)(p))

static __device__ __forceinline__ void wait_async_zero() {
#if HAVE_WAIT_ASYNC
  __builtin_amdgcn_s_wait_asynccnt(0);
#else
  asm volatile("s_wait_asynccnt 0" ::: "memory");
#endif
}

/* -------------------- phase B: streaming margin+softmax+scatter -----------
   Defined FIRST so the disasm snippet shows the hot kernel (async loads). */

__global__ void __launch_bounds__(THREADS) ism_main_kernel(
    const float* __restrict__ in, const int* __restrict__ labels,
    const int* __restrict__ pos, float* __restrict__ outData,
    float* __restrict__ outLabel, float* __restrict__ terms) {
  __shared__ float row[C_COLS];          /* 32 KB row tile */
  __shared__ float red[THREADS];
  const int r = blockIdx.x;
  const int t = threadIdx.x;
  const float* src = in + (size_t)r * C_COLS;

#if HAVE_ASYNC_B128
  {
    const char* g = (const char*)src + t * 16;
    char* l = (char*)row + t * 16;
    __builtin_amdgcn_global_load_async_to_lds_b128(AS1V(g), AS3V(l), 0, 0);
    __builtin_amdgcn_global_load_async_to_lds_b128(AS1V(g), AS3V(l), 4096, 0);
    __builtin_amdgcn_global_load_async_to_lds_b128(AS1V(g), AS3V(l), 8192, 0);
    __builtin_amdgcn_global_load_async_to_lds_b128(AS1V(g), AS3V(l), 12288, 0);
    __builtin_amdgcn_global_load_async_to_lds_b128(AS1V(g), AS3V(l), 16384, 0);
    __builtin_amdgcn_global_load_async_to_lds_b128(AS1V(g), AS3V(l), 20480, 0);
    __builtin_amdgcn_global_load_async_to_lds_b128(AS1V(g), AS3V(l), 24576, 0);
    __builtin_amdgcn_global_load_async_to_lds_b128(AS1V(g), AS3V(l), 28672, 0);
    wait_async_zero();
  }
#else
  {
    const fvec4* s4 = (const fvec4*)src;
    fvec4* r4 = (fvec4*)row;
#pragma unroll
    for (int i = 0; i < 8; ++i) r4[i * THREADS + t] = s4[i * THREADS + t];
  }
#endif
  __syncthreads();

  const int L = labels[r];
  if (t == 0) {
    const float v = row[L];
    row[L] = (v > 0.0f) ? (v / MARGIN_M - MARGIN_N) : (v * MARGIN_M - MARGIN_N);
  }
  __syncthreads();

  fvec4 v[8];
  fvec4* r4 = (fvec4*)row;
  float m = -INFINITY;
#pragma unroll
  for (int i = 0; i < 8; ++i) {
    v[i] = r4[i * THREADS + t];
    m = fmaxf(m, fmaxf(fmaxf(v[i].x, v[i].y), fmaxf(v[i].z, v[i].w)));
  }
  red[t] = m;
  __syncthreads();
  for (int off = 128; off > 0; off >>= 1) {
    if (t < off) red[t] = fmaxf(red[t], red[t + off]);
    __syncthreads();
  }
  m = red[0];
  __syncthreads();

  float s = 0.0f;
#pragma unroll
  for (int i = 0; i < 8; ++i)
    s += expf(v[i].x - m) + expf(v[i].y - m) + expf(v[i].z - m) + expf(v[i].w - m);
  red[t] = s;
  __syncthreads();
  for (int off = 128; off > 0; off >>= 1) {
    if (t < off) red[t] += red[t + off];
    __syncthreads();
  }
  const float lse = m + logf(red[0]);

  /* streaming scatter: written once, never re-read -> non-temporal stores */
  const int p = pos[r];
  fvec4* dst = (fvec4*)(outData + (size_t)p * C_COLS);
#pragma unroll
  for (int i = 0; i < 8; ++i)
    __builtin_nontemporal_store(v[i], &dst[i * THREADS + t]);

  if (t == 0) {
    outLabel[p] = (float)L;
    terms[r] = lse - row[L];  /* -(logp[target]) for this row */
  }
}

/* -------------------- phase A: permutation bookkeeping -------------------- */

__global__ void ism_init_kernel(int* __restrict__ first_occ) {
  int c = blockIdx.x * blockDim.x + threadIdx.x;
  if (c < C_COLS) first_occ[c] = 0x7FFFFFFF;
}

__global__ void __launch_bounds__(THREADS) ism_hist_kernel(
    const int* __restrict__ labels, int* __restrict__ first_occ,
    int* __restrict__ hist /* CHUNKS x C_COLS */) {
  __shared__ int h[C_COLS];
  const int chunk = blockIdx.x, t = threadIdx.x;
  for (int i = t; i < C_COLS; i += THREADS) h[i] = 0;
  __syncthreads();
  const int r = chunk * THREADS + t;
  const int L = labels[r];
  atomicMin(&first_occ[L], r);
  atomicAdd(&h[L], 1);
  __syncthreads();
  for (int i = t; i < C_COLS; i += THREADS) hist[(size_t)chunk * C_COLS + i] = h[i];
}

__global__ void ism_colprefix_kernel(int* __restrict__ hist, int* __restrict__ count) {
  const int c = blockIdx.x * blockDim.x + threadIdx.x;
  if (c >= C_COLS) return;
  int run = 0;
  for (int k = 0; k < CHUNKS; ++k) {
    const int v = hist[(size_t)k * C_COLS + c];
    hist[(size_t)k * C_COLS + c] = run;  /* exclusive prefix across chunks */
    run += v;
  }
  count[c] = run;
}

__global__ void ism_gval_kernel(const int* __restrict__ labels,
                                const int* __restrict__ first_occ,
                                const int* __restrict__ count,
                                int* __restrict__ gval) {
  const int r = blockIdx.x * blockDim.x + threadIdx.x;
  const int L = labels[r];
  gval[r] = (first_occ[L] == r) ? count[L] : 0;
}

/* exclusive prefix sum over B_ROWS ints: one block, 1024 threads x 16 elems */
__global__ void __launch_bounds__(1024) ism_scan_kernel(const int* __restrict__ g,
                                                        int* __restrict__ outp) {
  __shared__ int part[1024];
  const int t = threadIdx.x;
  const int base = t * 16;
  int vals[16];
  int s = 0;
#pragma unroll
  for (int i = 0; i < 16; ++i) { vals[i] = g[base + i]; s += vals[i]; }
  part[t] = s;
  __syncthreads();
  for (int off = 1; off < 1024; off <<= 1) {
    const int add = (t >= off) ? part[t - off] : 0;
    __syncthreads();
    part[t] += add;
    __syncthreads();
  }
  int run = (t > 0) ? part[t - 1] : 0;
#pragma unroll
  for (int i = 0; i < 16; ++i) { outp[base + i] = run; run += vals[i]; }
}

__global__ void __launch_bounds__(THREADS) ism_pos_kernel(
    const int* __restrict__ labels, const int* __restrict__ first_occ,
    const int* __restrict__ hist, const int* __restrict__ prefixG,
    int* __restrict__ pos) {
  __shared__ int lab[THREADS];
  const int chunk = blockIdx.x, t = threadIdx.x;
  const int r = chunk * THREADS + t;
  const int L = labels[r];
  lab[t] = L;
  __syncthreads();
  int inR = 0;
  for (int i = 0; i < t; ++i) inR += (lab[i] == L);
  pos[r] = prefixG[first_occ[L]] + hist[(size_t)chunk * C_COLS + L] + inR;
}

__global__ void __launch_bounds__(THREADS) ism_loss_kernel(
    const float* __restrict__ terms, float* __restrict__ lossOut) {
  __shared__ float acc[THREADS];
  const int t = threadIdx.x;
  float s = 0.0f;
  for (int i = t; i < B_ROWS; i += THREADS) s += terms[i];  /* fixed order */
  acc[t] = s;
  __syncthreads();
  for (int off = 128; off > 0; off >>= 1) {
    if (t < off) acc[t] += acc[t + off];
    __syncthreads();
  }
  if (t == 0) lossOut[0] = acc[0] / (float)B_ROWS;
}

/* ------------------------------- launcher --------------------------------- */

extern "C" void kernel_launch(void* const* d_in, const int* in_sizes, int n_in,
                              void* d_out, int out_size, void* d_ws, size_t ws_size,
                              hipStream_t stream) {
  (void)in_sizes; (void)n_in; (void)out_size; (void)ws_size;
  const float* logits = (const float*)d_in[0];
  const int*   labels = (const int*)d_in[1];

  char* ws = (char*)d_ws;
  int* hist      = (int*)(ws + 0);                 /* 64*8192*4 = 2097152 B */
  int* first_occ = (int*)(ws + 2097152);           /* 32768 B */
  int* count     = (int*)(ws + 2129920);           /* 32768 B */
  int* gval      = (int*)(ws + 2162688);           /* 65536 B */
  int* prefixG   = (int*)(ws + 2228224);           /* 65536 B */
  int* pos       = (int*)(ws + 2293760);           /* 65536 B */
  float* terms   = (float*)(ws + 2359296);         /* 65536 B */

  float* outData  = (float*)d_out;
  float* outLabel = outData + (size_t)B_ROWS * C_COLS;
  float* outLoss  = outLabel + B_ROWS;

  ism_init_kernel<<<C_COLS / THREADS, THREADS, 0, stream>>>(first_occ);
  ism_hist_kernel<<<CHUNKS, THREADS, 0, stream>>>(labels, first_occ, hist);
  ism_colprefix_kernel<<<C_COLS / THREADS, THREADS, 0, stream>>>(hist, count);
  ism_gval_kernel<<<B_ROWS / THREADS, THREADS, 0, stream>>>(labels, first_occ, count, gval);
  ism_scan_kernel<<<1, 1024, 0, stream>>>(gval, prefixG);
  ism_pos_kernel<<<CHUNKS, THREADS, 0, stream>>>(labels, first_occ, hist, prefixG, pos);
  ism_main_kernel<<<B_ROWS, THREADS, 0, stream>>>(logits, labels, pos, outData, outLabel, terms);
  ism_loss_kernel<<<1, THREADS, 0, stream>>>(terms, outLoss);
}